// RGCNEncoder_31705448579685
// MI455X (gfx1250) — compile-verified
//
#include <hip/hip_runtime.h>
#include <hip/hip_bf16.h>

// ---------------------------------------------------------------------------
// RGCN encoder for MI455X (gfx1250, wave32).
// Strategy: aggregate-then-transform (block-diag transform commutes with the
// per-(relation,dst) segment mean), so edge traffic touches only raw h rows
// and an 82 MB accumulator S[r][dst][H] that stays resident in the 192 MB L2.
// All GEMMs use V_WMMA_F32_16X16X4_F32 (full fp32 precision; the kernel is
// HBM/L2-bandwidth bound, so low-precision WMMA would not help).
// ---------------------------------------------------------------------------

typedef __attribute__((ext_vector_type(2))) float v2f;
typedef __attribute__((ext_vector_type(8))) float v8f;

#define WMMA_F32X4(acc, a, b) \
    __builtin_amdgcn_wmma_f32_16x16x4_f32(false, (a), false, (b), (short)0, (acc), false, false)

// ---------------------------------------------------------------------------
// out[rowbase+16, H=128] = relu?( A[16xK] @ W[KxH] + bias + agg )
// 256 threads = 8 waves; wave w owns the 16x16 tile at columns [16w, 16w+16).
// A tile staged in LDS (padded stride to dodge 64-bank conflicts).
// ---------------------------------------------------------------------------
__global__ void __launch_bounds__(256)
fused_gemm_out(const float* __restrict__ A, const float* __restrict__ W,
               const float* __restrict__ bias, const float* __restrict__ agg,
               float* __restrict__ out, int K, int Nagg, int doRelu)
{
    const int H = 128;
    __shared__ float As[16 * 132];
    const int ldAs = K + 4;
    const int rowbase = blockIdx.x * 16;
    const int tid = threadIdx.x;

    for (int idx = tid; idx < 16 * K; idx += 256) {
        int r = idx / K, c0 = idx - r * K;
        As[r * ldAs + c0] = A[(size_t)(rowbase + r) * K + c0];
    }
    __syncthreads();

    const int wave = tid >> 5;
    const int lane = tid & 31;
    const int lanelo = lane & 15;
    const int lanehi = lane >> 4;
    const int colbase = wave * 16;
    const int col = colbase + lanelo;

    v8f acc = {};
    #pragma unroll 8
    for (int kk = 0; kk < K; kk += 4) {
        const int k0 = kk + 2 * lanehi;
        v2f a, b;
        a[0] = As[lanelo * ldAs + k0];
        a[1] = As[lanelo * ldAs + k0 + 1];
        b[0] = W[(size_t)k0 * H + col];
        b[1] = W[(size_t)(k0 + 1) * H + col];
        acc = WMMA_F32X4(acc, a, b);
    }

    const float bv = bias ? bias[col] : 0.0f;
    #pragma unroll
    for (int i = 0; i < 8; ++i) {
        const int row = rowbase + i + 8 * lanehi;
        float v = acc[i] + bv;
        if (agg && row < Nagg) v += agg[(size_t)row * H + col];
        if (doRelu) v = fmaxf(v, 0.0f);
        out[(size_t)row * H + col] = v;
    }
}

// ---------------------------------------------------------------------------
// One wave per edge: S[r][dst][:] += h[src][:]; cnt[r][dst] += 1.
// 4 coalesced f32 atomics per lane; S + cnt are L2-resident (83 MB < 192 MB).
// ---------------------------------------------------------------------------
__global__ void __launch_bounds__(256)
edge_scatter(const float* __restrict__ h, const int* __restrict__ ei,
             const int* __restrict__ ea, int jcol, int E, int N,
             float* __restrict__ S, float* __restrict__ cnt)
{
    const int H = 128;
    const int gid = blockIdx.x * blockDim.x + threadIdx.x;
    const int e = gid >> 5;
    const int lane = gid & 31;
    if (e >= E) return;

    const int s = ei[e];                 // edge_index[0][e]
    const int d = ei[E + e];             // edge_index[1][e]
    const int r = ea[(size_t)e * 3 + jcol];

    const float* __restrict__ hs = h + (size_t)s * H;
    float* __restrict__ Sp = S + ((size_t)r * N + d) * H;
    #pragma unroll
    for (int q = 0; q < 4; ++q) {
        const int c0 = lane + 32 * q;
        atomicAdd(&Sp[c0], hs[c0]);
    }
    if (lane == 0) atomicAdd(&cnt[(size_t)r * N + d], 1.0f);
}

// ---------------------------------------------------------------------------
// agg[n][:] = sum_r ( S[r][n] (block-diag) @ W[r] ) / max(cnt[r][n], 1)
// Grid: N/16 row tiles. 8 waves cover H=128; wave w -> diag block b=w/2,
// sub-columns (w&1)*16 + lane%16.
// Per relation: the 16x128 row-tile of S[r] is staged into padded LDS with
// fully coalesced 128B/wave global loads (256 thr x 8 contiguous floats),
// then WMMA A-fragments are fed from LDS (ds_load path). cnt is staged too.
// 8 relations x 8 K-steps = 64 WMMAs/wave.
// ---------------------------------------------------------------------------
__global__ void __launch_bounds__(256)
rel_block_transform(const float* __restrict__ S, const float* __restrict__ cnt,
                    const float* __restrict__ Wij, float* __restrict__ agg, int N)
{
    const int H = 128, Bb = 4, c = 32, ldT = 132;
    __shared__ float Ts[16 * ldT];      // 16 rows x 128 cols, padded stride
    __shared__ float cnts[16];
    const int rowbase = blockIdx.x * 16;
    const int tid = threadIdx.x;
    const int wave = tid >> 5;
    const int lane = tid & 31;
    const int lanelo = lane & 15;
    const int lanehi = lane >> 4;
    const int b = wave >> 1;                       // diag block 0..3
    const int colsub = (wave & 1) * 16 + lanelo;   // 0..31 within block
    const int col = b * 32 + colsub;

    // cooperative-load indices: 8 contiguous floats per thread per tile
    const int ldRow = tid >> 4;            // 0..15
    const int ldCol = (tid & 15) * 8;      // 0,8,...,120

    v8f aggacc = {};
    for (int r = 0; r < 8; ++r) {
        __syncthreads();   // previous tile fully consumed before overwrite
        {
            const float* __restrict__ Sg =
                S + ((size_t)r * N + rowbase + ldRow) * H + ldCol;
            float* __restrict__ Td = Ts + ldRow * ldT + ldCol;
            #pragma unroll
            for (int q = 0; q < 8; ++q) Td[q] = Sg[q];
            if (tid < 16) cnts[tid] = cnt[(size_t)r * N + rowbase + tid];
        }
        __syncthreads();

        const float* __restrict__ Wr = Wij + ((size_t)r * Bb + b) * c * c;
        v8f acc = {};
        #pragma unroll
        for (int kk = 0; kk < 32; kk += 4) {
            const int k0 = kk + 2 * lanehi;
            v2f a, bv;
            a[0] = Ts[lanelo * ldT + b * 32 + k0];
            a[1] = Ts[lanelo * ldT + b * 32 + k0 + 1];
            bv[0] = Wr[k0 * c + colsub];
            bv[1] = Wr[(k0 + 1) * c + colsub];
            acc = WMMA_F32X4(acc, a, bv);
        }
        #pragma unroll
        for (int i = 0; i < 8; ++i) {
            const float cv = cnts[i + 8 * lanehi];
            aggacc[i] += acc[i] * (1.0f / fmaxf(cv, 1.0f));
        }
    }
    #pragma unroll
    for (int i = 0; i < 8; ++i)
        agg[(size_t)(rowbase + i + 8 * lanehi) * H + col] = aggacc[i];
}

// ---------------------------------------------------------------------------
extern "C" void kernel_launch(void* const* d_in, const int* in_sizes, int n_in,
                              void* d_out, int out_size, void* d_ws, size_t ws_size,
                              hipStream_t stream)
{
    (void)in_sizes; (void)n_in; (void)out_size; (void)ws_size;
    const int N = 20000, E = 640000, F = 64, H = 128, R = 8;

    const float* x    = (const float*)d_in[0];
    const int*   ei   = (const int*)  d_in[1];
    const int*   ea   = (const int*)  d_in[2];
    const float* femb = (const float*)d_in[3];
    const float* cw   = (const float*)d_in[4];   // (2,3,8,4,32,32)
    const float* cr   = (const float*)d_in[5];   // (2,3,128,128)
    const float* cb   = (const float*)d_in[6];   // (2,3,128)
    float* out = (float*)d_out;                  // (9N, 128)

    float* ws  = (float*)d_ws;
    float* h0  = ws;                             // N*H
    float* h1  = h0  + (size_t)N * H;            // 3N*H
    float* S   = h1  + (size_t)3 * N * H;        // R*N*H
    float* cnt = S   + (size_t)R * N * H;        // R*N   (contiguous after S)
    float* agg = cnt + (size_t)R * N;            // N*H

    // h0 = x @ feature_emb   (20000x64 @ 64x128)
    fused_gemm_out<<<N / 16, 256, 0, stream>>>(x, femb, nullptr, nullptr, h0,
                                               F, /*Nagg=*/0, /*relu=*/0);

    for (int layer = 0; layer < 2; ++layer) {
        const float* hin  = (layer == 0) ? h0 : h1;
        const int    Min  = (layer == 0) ? N : 3 * N;
        float*       hout = (layer == 0) ? h1 : out;
        for (int j = 0; j < 3; ++j) {
            // zero S and cnt (contiguous)
            hipMemsetAsync(S, 0, ((size_t)R * N * H + (size_t)R * N) * sizeof(float), stream);
            // per-edge scatter of raw features into per-relation sums
            edge_scatter<<<(E * 32) / 256, 256, 0, stream>>>(hin, ei, ea, j, E, N, S, cnt);
            // dense per-relation block-diag transform + mean + sum over relations
            rel_block_transform<<<N / 16, 256, 0, stream>>>(
                S, cnt, cw + (size_t)(layer * 3 + j) * R * 4 * 32 * 32, agg, N);
            // out = relu(agg + hin @ root + bias), stored at concat offset
            fused_gemm_out<<<Min / 16, 256, 0, stream>>>(
                hin, cr + (size_t)(layer * 3 + j) * H * H,
                cb + (size_t)(layer * 3 + j) * H, agg,
                hout + (size_t)j * Min * H, H, /*Nagg=*/N, /*relu=*/1);
        }
    }
}